// VGGTCrossFrameRKDAngleLoss_36524401885588
// MI455X (gfx1250) — compile-verified
//
#include <hip/hip_runtime.h>
#include <hip/hip_bf16.h>

typedef __attribute__((ext_vector_type(2))) float f32x2;
typedef __attribute__((ext_vector_type(8))) float f32x8;

// ---------------- problem constants ----------------
// teacher (B,8,P,D), student (B,4,P,D), B=2, P=2048, D=1024
// num_ref = num_shared = 128, TOPK=4, extra frames {1,3,5,7} -> E = 4*2048 = 8192
#define NB   2
#define NP   2048
#define ND   1024
#define NR   128
#define NS   128
#define NE   8192
#define NK   4
#define NPAIR 3

// ---------------- workspace layout (float units) ----------------
constexpr long REF_T   = 0;                          // B*NR*D        = 262144
constexpr long REF_S   = REF_T   + 262144;
constexpr long SH_T    = REF_S   + 262144;           // 3*B*NS*D      = 786432
constexpr long SH_S    = SH_T    + 786432;
constexpr long H_BUF   = SH_S    + 786432;           // B*NR*4*D      = 1048576
constexpr long SIM     = H_BUF   + 1048576;          // B*NR*E        = 2097152
constexpr long RSH_T   = SIM     + 2097152;          // 3*B*NR*NS     = 98304
constexpr long RSH_S   = RSH_T   + 98304;
constexpr long HSH_T   = RSH_S   + 98304;            // 3*B*NR*4*NS   = 393216
constexpr long HSH_S   = HSH_T   + 393216;
constexpr long RN2_T   = HSH_S   + 393216;           // 256
constexpr long RN2_S   = RN2_T   + 256;
constexpr long RINV_T  = RN2_S   + 256;
constexpr long SHN2_T  = RINV_T  + 256;              // 768
constexpr long SHN2_S  = SHN2_T  + 768;
constexpr long EINV    = SHN2_S  + 768;              // B*E = 16384
constexpr long HN2     = EINV    + 16384;            // 1024
constexpr long RH_T    = HN2     + 1024;
constexpr long RH_S    = RH_T    + 1024;
constexpr long PARTIAL = RH_S    + 1024;             // 1536 (rounded 2048)
constexpr long TOPK_O  = PARTIAL + 2048;             // 1024 ints
// total = TOPK_O + 1024 floats = 6,250,752 floats ~= 25.0 MB

// ---------------- helpers ----------------
__device__ __forceinline__ float block_reduce_sum(float v) {
  // blockDim.x == 256 (8 waves of 32)
  __shared__ float red[8];
  __syncthreads();                      // protect 'red' across repeated calls
  #pragma unroll
  for (int off = 16; off > 0; off >>= 1) v += __shfl_down(v, off, 32);
  const int lane = threadIdx.x & 31;
  const int w    = threadIdx.x >> 5;
  if (lane == 0) red[w] = v;
  __syncthreads();
  float r = 0.f;
  if (w == 0) {
    r = (lane < 8) ? red[lane] : 0.f;
    #pragma unroll
    for (int off = 4; off > 0; off >>= 1) r += __shfl_down(r, off, 32);
  }
  return r;                             // valid in thread 0
}

__device__ __forceinline__ float huber(float e) {
  float ae = fabsf(e);
  return (ae <= 1.0f) ? 0.5f * e * e : (ae - 0.5f);
}

__device__ __forceinline__ void angles(float rn2, float hn2, float shn2,
                                       float rh, float rsh, float hsh,
                                       float& a1, float& a2, float& a3) {
  const float EPSF = 1e-8f;
  float nrsh = fmaxf(sqrtf(fmaxf(rn2 + shn2 - 2.f * rsh, 0.f)), EPSF);
  float nrh  = fmaxf(sqrtf(fmaxf(rn2 + hn2  - 2.f * rh,  0.f)), EPSF);
  float nhsh = fmaxf(sqrtf(fmaxf(hn2 + shn2 - 2.f * hsh, 0.f)), EPSF);
  a1 = (hsh - rsh - rh + rn2)  / (nrsh * nrh);
  a2 = (rsh - rh - hsh + hn2)  / (nrh * nhsh);
  a3 = (rh - rsh - hsh + shn2) / (nrsh * nhsh);
}

// ---------------- K1: gather rows + squared norms ----------------
// grid 2048 blocks x 256 threads.
//   rid <  512 : ref rows   (side 0 = teacher frame0, side 1 = student frame0)
//   rid >= 512 : shared rows for pair p, side t/s
__global__ void gather_rows(const float* __restrict__ teacher,
                            const float* __restrict__ student,
                            const int* __restrict__ ref_perm,
                            const int* __restrict__ shared_perm,
                            float* __restrict__ ws) {
  const int rid = blockIdx.x;
  const float* src;
  float* dst;
  float* n2out;
  float* invout = nullptr;
  if (rid < 512) {
    const int side = rid >> 8;
    const int j = rid & 255;         // b*128 + r
    const int b = j >> 7;
    const int r = j & 127;
    const int p = ref_perm[r];
    if (side == 0) {
      src   = teacher + (((long)b * 8 + 0) * NP + p) * ND;
      dst   = ws + REF_T + (long)j * ND;
      n2out = ws + RN2_T + j;
      invout = ws + RINV_T + j;
    } else {
      src   = student + (((long)b * 4 + 0) * NP + p) * ND;
      dst   = ws + REF_S + (long)j * ND;
      n2out = ws + RN2_S + j;
    }
  } else {
    const int rid2 = rid - 512;
    const int pair = rid2 / 512;
    const int rem  = rid2 & 511;
    const int side = rem >> 8;
    const int j = rem & 255;         // b*128 + s
    const int b = j >> 7;
    const int s = j & 127;
    const int p = shared_perm[s];
    const int TIDX[NPAIR] = {2, 4, 6};
    const int SIDX[NPAIR] = {1, 2, 3};
    if (side == 0) {
      src   = teacher + (((long)b * 8 + TIDX[pair]) * NP + p) * ND;
      dst   = ws + SH_T + ((long)pair * 256 + j) * ND;
      n2out = ws + SHN2_T + pair * 256 + j;
    } else {
      src   = student + (((long)b * 4 + SIDX[pair]) * NP + p) * ND;
      dst   = ws + SH_S + ((long)pair * 256 + j) * ND;
      n2out = ws + SHN2_S + pair * 256 + j;
    }
  }
  float ss = 0.f;
  for (int d = threadIdx.x; d < ND; d += 256) {
    float x = src[d];
    dst[d] = x;
    ss += x * x;
  }
  float tot = block_reduce_sum(ss);
  if (threadIdx.x == 0) {
    *n2out = tot;
    if (invout) *invout = 1.f / fmaxf(sqrtf(tot), 1e-12f);
  }
}

// ---------------- K2: inverse norms of extra rows ----------------
__global__ void extra_norms(const float* __restrict__ teacher, float* __restrict__ einv) {
  const int blk = blockIdx.x;        // b*8192 + e
  const int b = blk >> 13;
  const int e = blk & (NE - 1);
  const int frame = 1 + 2 * (e >> 11);
  const int p = e & (NP - 1);
  const float* src = teacher + (((long)b * 8 + frame) * NP + p) * ND;
  float ss = 0.f;
  for (int d = threadIdx.x; d < ND; d += 256) { float x = src[d]; ss += x * x; }
  float tot = block_reduce_sum(ss);
  if (threadIdx.x == 0) einv[blk] = 1.f / fmaxf(sqrtf(tot), 1e-12f);
}

// ---------------- K3: similarity GEMM via f32 WMMA ----------------
// sim[b][m][e] = (ref_t . extra) * rinv[m] * einv[e]; one wave per 16x16 tile.
__global__ void sim_gemm(const float* __restrict__ teacher, float* __restrict__ ws) {
  const int nt = blockIdx.x;         // 0..511 (E tiles)
  const int mt = blockIdx.y;         // 0..7   (NR tiles)
  const int b  = blockIdx.z;
  const int lane = threadIdx.x;
  const int row = lane & 15;
  const int hi  = lane >> 4;
  const int e0 = nt * 16;
  const int frame = 1 + 2 * (e0 >> 11);   // each 16-row tile stays in one frame
  const int p0 = e0 & (NP - 1);
  const float* A = ws + REF_T + ((long)b * NR + mt * 16 + row) * ND + 2 * hi;
  const float* Y = teacher + (((long)b * 8 + frame) * NP + p0 + row) * ND + 2 * hi;
  f32x8 acc = {};
  #pragma unroll 8
  for (int k = 0; k < ND; k += 4) {
    f32x2 af = *(const f32x2*)(A + k);
    f32x2 bf = *(const f32x2*)(Y + k);
    acc = __builtin_amdgcn_wmma_f32_16x16x4_f32(false, af, false, bf,
                                                (short)0, acc, false, false);
  }
  const float ei = ws[EINV + b * NE + e0 + row];
  float* simout = ws + SIM + ((long)b * NR + mt * 16) * NE + e0 + row;
  #pragma unroll
  for (int v = 0; v < 8; ++v) {
    const int m = v + 8 * hi;
    simout[(long)m * NE] = acc[v] * ws[RINV_T + b * NR + mt * 16 + m] * ei;
  }
}

// ---------------- generic C = A . Y^T GEMM (f32 WMMA), batched ----------------
// A: (b, M, 1024) rows, Y: (b, N, 1024) rows, C: (b, M, ldC)
__global__ void gemm_xyt(const float* __restrict__ A, long sA,
                         const float* __restrict__ Y, long sY,
                         float* __restrict__ C, long sC, int ldC, int K) {
  const int nt = blockIdx.x, mt = blockIdx.y, b = blockIdx.z;
  const int lane = threadIdx.x;
  const int row = lane & 15;
  const int hi  = lane >> 4;
  const float* Ap = A + (long)b * sA + (long)(mt * 16 + row) * ND + 2 * hi;
  const float* Yp = Y + (long)b * sY + (long)(nt * 16 + row) * ND + 2 * hi;
  f32x8 acc = {};
  #pragma unroll 8
  for (int k = 0; k < K; k += 4) {
    f32x2 af = *(const f32x2*)(Ap + k);
    f32x2 bf = *(const f32x2*)(Yp + k);
    acc = __builtin_amdgcn_wmma_f32_16x16x4_f32(false, af, false, bf,
                                                (short)0, acc, false, false);
  }
  float* Cp = C + (long)b * sC + (long)(mt * 16) * ldC + nt * 16 + row;
  #pragma unroll
  for (int v = 0; v < 8; ++v)
    Cp[(long)(v + 8 * hi) * ldC] = acc[v];
}

// ---------------- K4: top-4 per sim row (ties -> lowest index) ----------------
__device__ __forceinline__ void tk_insert(float* bv, int* bi, float x, int e) {
  if (x > bv[3] || (x == bv[3] && e < bi[3])) {
    bv[3] = x; bi[3] = e;
    #pragma unroll
    for (int j = 3; j > 0; --j) {
      bool sw = (bv[j] > bv[j - 1]) || (bv[j] == bv[j - 1] && bi[j] < bi[j - 1]);
      if (sw) {
        float tv = bv[j]; bv[j] = bv[j - 1]; bv[j - 1] = tv;
        int ti = bi[j]; bi[j] = bi[j - 1]; bi[j - 1] = ti;
      }
    }
  }
}

__global__ void topk_kernel(const float* __restrict__ sim, int* __restrict__ topk) {
  const int rowid = blockIdx.x;      // b*128 + r
  const float* p = sim + (long)rowid * NE;
  float bv[NK] = {-3.4e38f, -3.4e38f, -3.4e38f, -3.4e38f};
  int   bi[NK] = {0x7FFFFFFF, 0x7FFFFFFF, 0x7FFFFFFF, 0x7FFFFFFF};
  for (int e = threadIdx.x; e < NE; e += 256) tk_insert(bv, bi, p[e], e);
  __shared__ float sv[256 * NK];
  __shared__ int   si[256 * NK];
  #pragma unroll
  for (int j = 0; j < NK; ++j) {
    sv[threadIdx.x * NK + j] = bv[j];
    si[threadIdx.x * NK + j] = bi[j];
  }
  __syncthreads();
  if (threadIdx.x == 0) {
    float ov[NK] = {-3.4e38f, -3.4e38f, -3.4e38f, -3.4e38f};
    int   oi[NK] = {0x7FFFFFFF, 0x7FFFFFFF, 0x7FFFFFFF, 0x7FFFFFFF};
    for (int c = 0; c < 256 * NK; ++c) tk_insert(ov, oi, sv[c], si[c]);
    #pragma unroll
    for (int j = 0; j < NK; ++j) topk[rowid * NK + j] = oi[j];
  }
}

// ---------------- K5: gather H + Hn2 + RH dots ----------------
__global__ void gather_H(const float* __restrict__ teacher,
                         const int* __restrict__ topk,
                         float* __restrict__ ws) {
  const int blk = blockIdx.x;        // b*512 + r*4 + k
  const int b = blk >> 9;
  const int e = topk[blk];
  const int frame = 1 + 2 * (e >> 11);
  const int p = e & (NP - 1);
  const int r = (blk >> 2) & 127;
  const float* src = teacher + (((long)b * 8 + frame) * NP + p) * ND;
  float* dst = ws + H_BUF + (long)blk * ND;
  const float* rt = ws + REF_T + ((long)b * NR + r) * ND;
  const float* rs = ws + REF_S + ((long)b * NR + r) * ND;
  float shh = 0.f, srt = 0.f, srs = 0.f;
  for (int d = threadIdx.x; d < ND; d += 256) {
    float h = src[d];
    dst[d] = h;
    shh += h * h;
    srt += h * rt[d];
    srs += h * rs[d];
  }
  float a  = block_reduce_sum(shh);
  float c  = block_reduce_sum(srt);
  float dd = block_reduce_sum(srs);
  if (threadIdx.x == 0) {
    ws[HN2  + blk] = a;
    ws[RH_T + blk] = c;
    ws[RH_S + blk] = dd;
  }
}

// ---------------- K8: angles + Huber + per-block partial sums ----------------
__global__ void angle_loss(const float* __restrict__ ws, float* __restrict__ partial) {
  long t = (long)blockIdx.x * 256 + threadIdx.x;   // exactly 3*2*128*128*4 threads
  const int k4 = (int)(t & 3);  t >>= 2;
  const int s  = (int)(t & 127); t >>= 7;
  const int r  = (int)(t & 127); t >>= 7;
  const int b  = (int)(t & 1);
  const int pair = (int)(t >> 1);
  const int br  = b * NR + r;
  const int brk = br * NK + k4;
  const int pbs = pair * 256 + b * NS + s;
  const float rn2t  = ws[RN2_T + br],  rn2s  = ws[RN2_S + br];
  const float hn2v  = ws[HN2 + brk];
  const float rht   = ws[RH_T + brk], rhs   = ws[RH_S + brk];
  const float shn2t = ws[SHN2_T + pbs], shn2s = ws[SHN2_S + pbs];
  const long rshoff = (long)pair * (NB * NR * NS) + (long)br * NS + s;
  const float rsht = ws[RSH_T + rshoff], rshs = ws[RSH_S + rshoff];
  const long hshoff = (long)pair * (NB * NR * NK * NS) + ((long)b * NR * NK + r * NK + k4) * NS + s;
  const float hsht = ws[HSH_T + hshoff], hshs = ws[HSH_S + hshoff];
  float a1t, a2t, a3t, a1s, a2s, a3s;
  angles(rn2t, hn2v, shn2t, rht, rsht, hsht, a1t, a2t, a3t);
  angles(rn2s, hn2v, shn2s, rhs, rshs, hshs, a1s, a2s, a3s);
  float contrib = huber(a1s - a1t) + huber(a2s - a2t) + huber(a3s - a3t);
  float sum = block_reduce_sum(contrib);
  if (threadIdx.x == 0) partial[blockIdx.x] = sum;
}

__global__ void final_reduce(const float* __restrict__ partial, float* __restrict__ out) {
  float s = 0.f;
  for (int i = threadIdx.x; i < 1536; i += 256) s += partial[i];
  float tot = block_reduce_sum(s);
  if (threadIdx.x == 0) out[0] = tot / 393216.0f;   // 3*B*NR*NS*TOPK
}

// ---------------- host launch ----------------
extern "C" void kernel_launch(void* const* d_in, const int* in_sizes, int n_in,
                              void* d_out, int out_size, void* d_ws, size_t ws_size,
                              hipStream_t stream) {
  (void)in_sizes; (void)n_in; (void)out_size; (void)ws_size;
  const float* teacher = (const float*)d_in[0];     // (2,8,2048,1024) f32
  const float* student = (const float*)d_in[1];     // (2,4,2048,1024) f32
  const int* ref_perm    = (const int*)d_in[2];     // 128 i32
  const int* shared_perm = (const int*)d_in[3];     // 128 i32
  float* out = (float*)d_out;
  float* ws  = (float*)d_ws;                        // needs ~25 MB
  int* topk  = (int*)(ws + TOPK_O);

  // 1) gathers + row norms
  gather_rows<<<2048, 256, 0, stream>>>(teacher, student, ref_perm, shared_perm, ws);
  // 2) extra row inverse norms
  extra_norms<<<NB * NE, 256, 0, stream>>>(teacher, ws + EINV);
  // 3) normalized similarity (WMMA f32)
  sim_gemm<<<dim3(NE / 16, NR / 16, NB), 32, 0, stream>>>(teacher, ws);
  // 4) top-4 indices per reference row
  topk_kernel<<<NB * NR, 256, 0, stream>>>(ws + SIM, topk);
  // 5) gather H, Hn2, RH dots
  gather_H<<<NB * NR * NK, 256, 0, stream>>>(teacher, topk, ws);
  // 6) RSh / HSh GEMMs (WMMA f32) per shared pair, teacher & student sides
  for (int pair = 0; pair < NPAIR; ++pair) {
    const float* sht = ws + SH_T + (long)pair * (NB * NS * ND);
    const float* shs = ws + SH_S + (long)pair * (NB * NS * ND);
    // RSh: (128 x 128) . K=1024
    gemm_xyt<<<dim3(NS / 16, NR / 16, NB), 32, 0, stream>>>(
        ws + REF_T, (long)NR * ND, sht, (long)NS * ND,
        ws + RSH_T + (long)pair * (NB * NR * NS), (long)NR * NS, NS, ND);
    gemm_xyt<<<dim3(NS / 16, NR / 16, NB), 32, 0, stream>>>(
        ws + REF_S, (long)NR * ND, shs, (long)NS * ND,
        ws + RSH_S + (long)pair * (NB * NR * NS), (long)NR * NS, NS, ND);
    // HSh: (512 x 128) . K=1024
    gemm_xyt<<<dim3(NS / 16, (NR * NK) / 16, NB), 32, 0, stream>>>(
        ws + H_BUF, (long)NR * NK * ND, sht, (long)NS * ND,
        ws + HSH_T + (long)pair * (NB * NR * NK * NS), (long)NR * NK * NS, NS, ND);
    gemm_xyt<<<dim3(NS / 16, (NR * NK) / 16, NB), 32, 0, stream>>>(
        ws + H_BUF, (long)NR * NK * ND, shs, (long)NS * ND,
        ws + HSH_S + (long)pair * (NB * NR * NK * NS), (long)NR * NK * NS, NS, ND);
  }
  // 7) angle differences + Huber, deterministic two-stage reduction
  angle_loss<<<1536, 256, 0, stream>>>(ws, ws + PARTIAL);
  final_reduce<<<1, 256, 0, stream>>>(ws + PARTIAL, out);
}